// StatefulGatedAttention_103079215210
// MI455X (gfx1250) — compile-verified
//
#include <hip/hip_runtime.h>

// ---------------------------------------------------------------------------
// StatefulGatedAttention for MI455X (gfx1250, wave32)
//   - WMMA f32 16x16x4 for both GEMMs (workload is HBM-bound at ~400MB /
//     23.3 TB/s ~ 17us; fp32 matrix path keeps full precision at zero cost)
//   - GLOBAL_LOAD_ASYNC_TO_LDS_B32 + double-buffered LDS for GEMM staging
//   - chunked 3-pass parallel scan for the sequential recurrence
// ---------------------------------------------------------------------------

typedef __attribute__((ext_vector_type(2))) float v2f;
typedef __attribute__((ext_vector_type(8))) float v8f;

#define B_DIM 8
#define C_DIM 384
#define T_DIM 4096
#define NCH   32                 // scan chunks
#define CHLEN (T_DIM / NCH)      // 128 timesteps per chunk

__device__ __forceinline__ float fast_sigmoid(float x) {
    return 1.0f / (1.0f + __expf(-x));
}
__device__ __forceinline__ float fast_tanh(float x) {
    return 1.0f - 2.0f / (__expf(2.0f * x) + 1.0f);
}

// Async copy 4B from global (uniform SGPR base + 32-bit per-lane byte offset)
// directly into LDS, bypassing VGPRs. Tracked by ASYNCcnt.
__device__ __forceinline__ void async_copy_b32(const void* lds_ptr,
                                               const void* gbase,
                                               unsigned gvoff_bytes) {
    // generic shared pointer = {aperture_hi32, lds_byte_offset} -> truncate
    unsigned lds_addr = (unsigned)(size_t)lds_ptr;
    asm volatile("global_load_async_to_lds_b32 %0, %1, %2"
                 :
                 : "v"(lds_addr), "v"(gvoff_bytes), "s"(gbase)
                 : "memory");
}
__device__ __forceinline__ void wait_async0() {
    asm volatile("s_wait_asynccnt 0x0" ::: "memory");
}

// ---------------------------------------------------------------------------
// Kernel A: fused gate GEMM + activations.
// Block = 128 threads (4 waves), tile = 32(t) x 32(c); each wave accumulates
// a 16x16 subtile for each of the 3 gates so ic = sig(i)*tanh(cand) fuses in
// the epilogue. Writes F[b][t][c], IC[b][t][c] (B,T,C layout).
// WMMA mapping: M = t, N = c, K = c_in.  Double-buffered async LDS staging.
// ---------------------------------------------------------------------------
__global__ __launch_bounds__(128) void gate_gemm_act_kernel(
    const float* __restrict__ x,     // (B, C, T)
    const float* __restrict__ Wg,    // (3C, C)
    float* __restrict__ F,           // (B, T, C)
    float* __restrict__ IC)          // (B, T, C)
{
    __shared__ float lds_x[2][32 * 32];        // [k][t], t contiguous
    __shared__ float lds_w[2][3][32 * 33];     // [k][n] padded per gate

    const int tid  = threadIdx.x;
    const int lane = tid & 31;
    const int wave = tid >> 5;
    const int lm   = lane & 15;            // M/N index within 16
    const int koff = (lane >> 4) << 1;     // lanes 16-31 hold K+2
    const int roff = (lane >> 4) << 3;     // D: lanes 16-31 hold M = r+8
    const int wt   = (wave & 1) * 16;      // wave's t offset in tile
    const int wc   = (wave >> 1) * 16;     // wave's c offset in tile

    const int t0 = blockIdx.x * 32;
    const int c0 = blockIdx.y * 32;
    const int b  = blockIdx.z;

    const float* xb = x + (size_t)b * C_DIM * T_DIM;   // uniform -> SGPR base

    v8f accF = {}; v8f accI = {}; v8f accC = {};

    // issue async staging for k-chunk kc into buffer buf
    auto stage = [&](int kc, int buf) {
        {
            const int t  = tid & 31;
            const int kb = tid >> 5;           // 0..3
            #pragma unroll
            for (int kk = 0; kk < 32; kk += 4) {
                const int k = kb + kk;
                async_copy_b32(&lds_x[buf][k * 32 + t], xb,
                               (unsigned)(((kc + k) * T_DIM + t0 + t) * 4));
            }
        }
        {
            const int k  = tid & 31;
            const int nb = tid >> 5;           // 0..3
            #pragma unroll
            for (int nn = 0; nn < 32; nn += 4) {
                const int n = nb + nn;
                #pragma unroll
                for (int g = 0; g < 3; ++g) {
                    async_copy_b32(&lds_w[buf][g][k * 33 + n], Wg,
                        (unsigned)(((g * C_DIM + c0 + n) * C_DIM + kc + k) * 4));
                }
            }
        }
    };

    stage(0, 0);
    wait_async0();
    __syncthreads();

    int buf = 0;
    for (int kc = 0; kc < C_DIM; kc += 32) {
        if (kc + 32 < C_DIM) stage(kc + 32, buf ^ 1);   // overlap with compute

        #pragma unroll
        for (int k = 0; k < 32; k += 4) {
            v2f a, bf, bi, bc;
            a.x  = lds_x[buf][(k + koff)     * 32 + wt + lm];
            a.y  = lds_x[buf][(k + koff + 1) * 32 + wt + lm];
            bf.x = lds_w[buf][0][(k + koff)     * 33 + wc + lm];
            bf.y = lds_w[buf][0][(k + koff + 1) * 33 + wc + lm];
            bi.x = lds_w[buf][1][(k + koff)     * 33 + wc + lm];
            bi.y = lds_w[buf][1][(k + koff + 1) * 33 + wc + lm];
            bc.x = lds_w[buf][2][(k + koff)     * 33 + wc + lm];
            bc.y = lds_w[buf][2][(k + koff + 1) * 33 + wc + lm];
            accF = __builtin_amdgcn_wmma_f32_16x16x4_f32(
                       false, a, false, bf, (short)0, accF, false, false);
            accI = __builtin_amdgcn_wmma_f32_16x16x4_f32(
                       false, a, false, bi, (short)0, accI, false, false);
            accC = __builtin_amdgcn_wmma_f32_16x16x4_f32(
                       false, a, false, bc, (short)0, accC, false, false);
        }

        wait_async0();       // next-chunk copies complete
        __syncthreads();     // all waves done reading current buffer
        buf ^= 1;
    }

    // ---- epilogue: activations + store F / IC (lane = c -> contiguous) ----
    #pragma unroll
    for (int r = 0; r < 8; ++r) {
        const int t = t0 + wt + r + roff;
        const int c = c0 + wc + lm;
        const size_t idx = ((size_t)b * T_DIM + t) * C_DIM + c;
        F[idx]  = fast_sigmoid(accF[r]);
        IC[idx] = fast_sigmoid(accI[r]) * fast_tanh(accC[r]);
    }
}

// ---------------------------------------------------------------------------
// Scan pass 1: per-chunk aggregates (a = prod f, h = chunk-local scan end)
// ---------------------------------------------------------------------------
__global__ __launch_bounds__(128) void scan_pass1_kernel(
    const float* __restrict__ F, const float* __restrict__ IC,
    float* __restrict__ Aagg, float* __restrict__ Bagg)
{
    const int c  = blockIdx.x * 128 + threadIdx.x;   // 0..383
    const int ch = blockIdx.y;
    const int b  = blockIdx.z;
    const size_t base = ((size_t)b * T_DIM + (size_t)ch * CHLEN) * C_DIM + c;
    const float* Fp = F  + base;
    const float* Ip = IC + base;
    float a = 1.0f, h = 0.0f;
    for (int t = 0; t < CHLEN; ++t) {
        const float f = Fp[(size_t)t * C_DIM];
        h = __builtin_fmaf(f, h, Ip[(size_t)t * C_DIM]);
        a *= f;
    }
    const size_t o = ((size_t)b * NCH + ch) * C_DIM + c;
    Aagg[o] = a;
    Bagg[o] = h;
}

// ---------------------------------------------------------------------------
// Scan pass 2: tiny exclusive scan over the 32 chunks per (b, c).
// ---------------------------------------------------------------------------
__global__ __launch_bounds__(384) void scan_pass2_kernel(
    const float* __restrict__ Aagg, const float* __restrict__ Bagg,
    float* __restrict__ Init)
{
    const int c = threadIdx.x;   // 0..383
    const int b = blockIdx.x;
    float state = 0.0f;
    for (int ch = 0; ch < NCH; ++ch) {
        const size_t o = ((size_t)b * NCH + ch) * C_DIM + c;
        Init[o] = state;
        state = __builtin_fmaf(Aagg[o], state, Bagg[o]);
    }
}

// ---------------------------------------------------------------------------
// Scan pass 3: replay each chunk with correct initial state, emit all states.
// ---------------------------------------------------------------------------
__global__ __launch_bounds__(128) void scan_pass3_kernel(
    const float* __restrict__ F, const float* __restrict__ IC,
    const float* __restrict__ Init, float* __restrict__ S)
{
    const int c  = blockIdx.x * 128 + threadIdx.x;
    const int ch = blockIdx.y;
    const int b  = blockIdx.z;
    const size_t base = ((size_t)b * T_DIM + (size_t)ch * CHLEN) * C_DIM + c;
    const float* Fp = F  + base;
    const float* Ip = IC + base;
    float*       Sp = S  + base;
    float h = Init[((size_t)b * NCH + ch) * C_DIM + c];
    for (int t = 0; t < CHLEN; ++t) {
        h = __builtin_fmaf(Fp[(size_t)t * C_DIM], h, Ip[(size_t)t * C_DIM]);
        Sp[(size_t)t * C_DIM] = h;
    }
}

// ---------------------------------------------------------------------------
// Kernel C: output GEMM, out[b][o][t] = sum_c S[b][t][c] * Wo[o][c].
// WMMA mapping: M = o, N = t  -> D-fragment stores are contiguous along t
// (two 64B runs per store), matching the transposed (B,C,T) output layout.
// Double-buffered async LDS staging.
// ---------------------------------------------------------------------------
__global__ __launch_bounds__(128) void out_gemm_kernel(
    const float* __restrict__ S,     // (B, T, C)
    const float* __restrict__ Wo,    // (C, C)
    float* __restrict__ out)         // (B, C, T)
{
    __shared__ float lds_s[2][32 * 33];   // [k=c][t] padded
    __shared__ float lds_w[2][32 * 33];   // [k=c][o] padded

    const int tid  = threadIdx.x;
    const int lane = tid & 31;
    const int wave = tid >> 5;
    const int lm   = lane & 15;
    const int koff = (lane >> 4) << 1;
    const int roff = (lane >> 4) << 3;
    const int wt   = (wave & 1) * 16;    // t offset
    const int wo   = (wave >> 1) * 16;   // o offset

    const int t0 = blockIdx.x * 32;
    const int o0 = blockIdx.y * 32;
    const int b  = blockIdx.z;

    v8f acc = {};

    auto stage = [&](int kc, int buf) {
        const int cc = tid & 31;
        const int rb = tid >> 5;
        #pragma unroll
        for (int tt = 0; tt < 32; tt += 4) {
            const int t = rb + tt;
            async_copy_b32(&lds_s[buf][cc * 33 + t], S,
                (unsigned)((((b * T_DIM + t0 + t) * C_DIM) + kc + cc) * 4));
        }
        #pragma unroll
        for (int oo = 0; oo < 32; oo += 4) {
            const int o = rb + oo;
            async_copy_b32(&lds_w[buf][cc * 33 + o], Wo,
                (unsigned)((((o0 + o) * C_DIM) + kc + cc) * 4));
        }
    };

    stage(0, 0);
    wait_async0();
    __syncthreads();

    int buf = 0;
    for (int kc = 0; kc < C_DIM; kc += 32) {
        if (kc + 32 < C_DIM) stage(kc + 32, buf ^ 1);

        #pragma unroll
        for (int k = 0; k < 32; k += 4) {
            v2f a, bb;
            a.x  = lds_w[buf][(k + koff)     * 33 + wo + lm];   // A(m=o, k)
            a.y  = lds_w[buf][(k + koff + 1) * 33 + wo + lm];
            bb.x = lds_s[buf][(k + koff)     * 33 + wt + lm];   // B(k=c, n=t)
            bb.y = lds_s[buf][(k + koff + 1) * 33 + wt + lm];
            acc = __builtin_amdgcn_wmma_f32_16x16x4_f32(
                      false, a, false, bb, (short)0, acc, false, false);
        }

        wait_async0();
        __syncthreads();
        buf ^= 1;
    }

    // D: lane = n = t (contiguous), VGPR r -> m = o = r / r+8
    #pragma unroll
    for (int r = 0; r < 8; ++r) {
        const int o = o0 + wo + r + roff;
        const int t = t0 + wt + lm;
        out[((size_t)b * C_DIM + o) * T_DIM + t] = acc[r];
    }
}

// ---------------------------------------------------------------------------
// Host launcher
// ---------------------------------------------------------------------------
extern "C" void kernel_launch(void* const* d_in, const int* in_sizes, int n_in,
                              void* d_out, int out_size, void* d_ws, size_t ws_size,
                              hipStream_t stream) {
    (void)in_sizes; (void)n_in; (void)out_size; (void)ws_size;

    const float* x  = (const float*)d_in[0];   // (B, C, T)
    const float* Wg = (const float*)d_in[1];   // (3C, C)
    const float* Wo = (const float*)d_in[2];   // (C, C)
    float*       y  = (float*)d_out;           // (B, C, T)

    const size_t BTC  = (size_t)B_DIM * T_DIM * C_DIM;   // 12,582,912
    const size_t BNC  = (size_t)B_DIM * NCH * C_DIM;     // 98,304

    float* F    = (float*)d_ws;
    float* IC   = F    + BTC;
    float* S    = IC   + BTC;
    float* Aagg = S    + BTC;
    float* Bagg = Aagg + BNC;
    float* Init = Bagg + BNC;

    // 1) fused gate GEMM + activations
    {
        dim3 grid(T_DIM / 32, C_DIM / 32, B_DIM);   // 128 x 12 x 8
        gate_gemm_act_kernel<<<grid, 128, 0, stream>>>(x, Wg, F, IC);
    }
    // 2) chunked scan (3 passes)
    {
        dim3 grid1(C_DIM / 128, NCH, B_DIM);        // 3 x 32 x 8
        scan_pass1_kernel<<<grid1, 128, 0, stream>>>(F, IC, Aagg, Bagg);
        scan_pass2_kernel<<<B_DIM, C_DIM, 0, stream>>>(Aagg, Bagg, Init);
        scan_pass3_kernel<<<grid1, 128, 0, stream>>>(F, IC, Init, S);
    }
    // 3) output GEMM (writes transposed (B,C,T) directly)
    {
        dim3 grid(T_DIM / 32, C_DIM / 32, B_DIM);   // 128 x 12 x 8
        out_gemm_kernel<<<grid, 128, 0, stream>>>(S, Wo, y);
    }
}